// SpatialAttention_1726576853791
// MI455X (gfx1250) — compile-verified
//
#include <hip/hip_runtime.h>
#include <stdint.h>

// ---------------------------------------------------------------------------
// SpatialAttention for MI455X (gfx1250, wave32, WMMA + TDM).
//   0) pack assignment (N,N,R) -> Ap (2176 x 128) row-major
//   1) pre_kv[b] = Ap (2176x128) @ x[b] (128x512)         [WMMA f32, LDS-staged]
//   2) kv  = pre_kv (34816x512) @ kv_w^T (512x1024)       [WMMA f32 + TDM dbl-buf]
//   3) q   = x (2048x512) @ q_w^T                         [WMMA f32 + TDM dbl-buf]
//   4) attention: per (b,n,h) wave: logits(17)->softmax->weighted V
//   5) out = attn_out (2048x512) @ proj_w^T + proj_b      [WMMA f32 + TDM dbl-buf]
// ---------------------------------------------------------------------------

typedef __attribute__((ext_vector_type(2))) float v2f;
typedef __attribute__((ext_vector_type(8))) float v8f;
typedef __attribute__((ext_vector_type(4))) unsigned int u32x4;
typedef __attribute__((ext_vector_type(8))) int i32x8;
typedef __attribute__((ext_vector_type(4))) int i32x4;

#define BB 16
#define NN 128
#define CC 512
#define HH 8
#define RR 17
#define MR 2176      // N*R
#define MROWS 34816  // B*N*R
#define LPAD 34      // LDS row stride (32 data + 2 pad dwords)

// ---------------------------------------------------------------------------
// Kernel 0: Ap[m*17+r][n] = assignment[(m*128+n)*17 + r]
// ---------------------------------------------------------------------------
__global__ __launch_bounds__(256) void pack_assignment_k(
    const float* __restrict__ asg, float* __restrict__ Ap)
{
    int idx = blockIdx.x * 256 + threadIdx.x;
    if (idx >= MR * NN) return;
    int row = idx >> 7;    // m*17 + r
    int n   = idx & 127;
    int m   = row / RR;
    int r   = row - m * RR;
    Ap[idx] = asg[(m * NN + n) * RR + r];
}

// ---------------------------------------------------------------------------
// TDM: DMA a 64-row x 32-float fp32 tile (row stride = stride_elems) from
// global into LDS at lds_off_bytes, inserting 2 pad dwords after every 32
// dwords -> LDS row stride 34 floats (bank-conflict-free b64 fragment loads).
// D# built per CDNA5 ISA (group0/group1 bitfields).
// This toolchain exposes the 6-arg builtin:
//   (u32x4 g0, i32x8 g1, i32x4 g2, i32x4 g3, i32x8 ext, i32 cpol)
// groups 2/3 and the extension group are unused for a 2D tile -> zeros.
// ---------------------------------------------------------------------------
__device__ __forceinline__ void tdm_load_tile_64x32(
    const float* gsrc, unsigned int lds_off_bytes, unsigned int stride_elems)
{
    const unsigned long long ga = (unsigned long long)(uintptr_t)gsrc;
    u32x4 g0;
    g0[0] = 1u;                                     // count=1 (valid), user desc
    g0[1] = lds_off_bytes;                          // lds_addr [63:32]
    g0[2] = (unsigned int)ga;                       // global_addr [95:64]
    g0[3] = (unsigned int)((ga >> 32) & 0x01FFFFFFu) | (2u << 30); // addr[56:32], type=2

    i32x8 g1;
    // mask=0 | data_size=2 (4B) | pad_enable | pad_interval=4 (32dw) | pad_amount=1 (2dw)
    g1[0] = (int)((2u << 16) | (1u << 20) | (4u << 22) | (1u << 25));
    g1[1] = (int)(32u << 16);        // tensor_dim0 = 32 (bits 79:48)
    g1[2] = (int)(64u << 16);        // tensor_dim1 = 64 (bits 111:80)
    g1[3] = (int)(32u << 16);        // tile_dim0  = 32 (bits 127:112)
    g1[4] = (int)(64u);              // tile_dim1  = 64 (bits 143:128); tile_dim2 = 0
    g1[5] = (int)stride_elems;       // tensor_dim0_stride[31:0] (elements)
    g1[6] = 0;                       // stride hi / dim1_stride lo (unused, 2D)
    g1[7] = 0;
    i32x4 gz4; gz4[0] = 0; gz4[1] = 0; gz4[2] = 0; gz4[3] = 0;
    i32x8 gz8; gz8[0] = 0; gz8[1] = 0; gz8[2] = 0; gz8[3] = 0;
    gz8[4] = 0; gz8[5] = 0; gz8[6] = 0; gz8[7] = 0;
    __builtin_amdgcn_tensor_load_to_lds(g0, g1, gz4, gz4, gz8, 0);
}

// ---------------------------------------------------------------------------
// TDM-staged fp32 WMMA GEMM: C = A * W^T (+bias).  A: MxK row-major,
// W: NxK row-major (weights), C: MxN row-major.  M%64==N%64==K%32==0.
// Block = 128 thr (4 waves, 2x2) -> 64x64 C tile; wave = 32x32 (2x2 of 16x16).
// LDS double-buffered; wave 0 drives the TDM, one barrier per K-chunk.
// ---------------------------------------------------------------------------
template<bool WITH_BIAS>
__global__ __launch_bounds__(128) void gemm_wmma_f32_tdm_k(
    const float* __restrict__ A, const float* __restrict__ W,
    const float* __restrict__ bias, float* __restrict__ C,
    int K, int lda, int ldb, int ldc)
{
    __shared__ float lA[2][64 * LPAD];
    __shared__ float lB[2][64 * LPAD];

    const int tid  = threadIdx.x;
    const int lane = tid & 31;
    const int wave = tid >> 5;
    const int wm   = wave >> 1;
    const int wn   = wave & 1;
    const int half = lane >> 4;      // K-pair select (ISA A/B frag layout)
    const int l16  = lane & 15;
    const int row0 = blockIdx.y * 64;
    const int col0 = blockIdx.x * 64;

    v8f acc00 = {}, acc01 = {}, acc10 = {}, acc11 = {};

    const unsigned int ldsA0 = (unsigned int)(uintptr_t)(&lA[0][0]);
    const unsigned int ldsA1 = (unsigned int)(uintptr_t)(&lA[1][0]);
    const unsigned int ldsB0 = (unsigned int)(uintptr_t)(&lB[0][0]);
    const unsigned int ldsB1 = (unsigned int)(uintptr_t)(&lB[1][0]);
    const int nch = K >> 5;

    if (wave == 0) {
        tdm_load_tile_64x32(A + (size_t)row0 * lda, ldsA0, (unsigned int)lda);
        tdm_load_tile_64x32(W + (size_t)col0 * ldb, ldsB0, (unsigned int)ldb);
    }

    int cur = 0;
    for (int ic = 0; ic < nch; ++ic) {
        if (wave == 0) __builtin_amdgcn_s_wait_tensorcnt(0);
        __syncthreads();                        // buf[cur] ready; buf[cur^1] free
        if (wave == 0 && ic + 1 < nch) {        // prefetch next chunk via TDM
            const int kc = (ic + 1) << 5;
            tdm_load_tile_64x32(A + (size_t)row0 * lda + kc,
                                cur ? ldsA0 : ldsA1, (unsigned int)lda);
            tdm_load_tile_64x32(W + (size_t)col0 * ldb + kc,
                                cur ? ldsB0 : ldsB1, (unsigned int)ldb);
        }
        const float* pa = lA[cur];
        const float* pb = lB[cur];
        const int ra0 = (wm * 32 + l16) * LPAD;
        const int ra1 = ra0 + 16 * LPAD;
        const int cb0 = (wn * 32 + l16) * LPAD;
        const int cb1 = cb0 + 16 * LPAD;
        #pragma unroll
        for (int kk = 0; kk < 32; kk += 4) {
            const int ko = kk + 2 * half;
            const float2 a0f = *reinterpret_cast<const float2*>(&pa[ra0 + ko]);
            const float2 a1f = *reinterpret_cast<const float2*>(&pa[ra1 + ko]);
            const float2 b0f = *reinterpret_cast<const float2*>(&pb[cb0 + ko]);
            const float2 b1f = *reinterpret_cast<const float2*>(&pb[cb1 + ko]);
            v2f a0; a0.x = a0f.x; a0.y = a0f.y;
            v2f a1; a1.x = a1f.x; a1.y = a1f.y;
            v2f b0; b0.x = b0f.x; b0.y = b0f.y;
            v2f b1; b1.x = b1f.x; b1.y = b1f.y;
            acc00 = __builtin_amdgcn_wmma_f32_16x16x4_f32(false, a0, false, b0, (short)0, acc00, false, false);
            acc01 = __builtin_amdgcn_wmma_f32_16x16x4_f32(false, a0, false, b1, (short)0, acc01, false, false);
            acc10 = __builtin_amdgcn_wmma_f32_16x16x4_f32(false, a1, false, b0, (short)0, acc10, false, false);
            acc11 = __builtin_amdgcn_wmma_f32_16x16x4_f32(false, a1, false, b1, (short)0, acc11, false, false);
        }
        cur ^= 1;
    }

    // epilogue (32-bit indexing; all C buffers < 2^31 elements)
    const int ccol0 = col0 + wn * 32 + l16;
    const int ccol1 = ccol0 + 16;
    float bv0 = 0.0f, bv1 = 0.0f;
    if (WITH_BIAS) { bv0 = bias[ccol0]; bv1 = bias[ccol1]; }
    const int crow0 = row0 + wm * 32 + 8 * half;
    const int crow1 = crow0 + 16;
    #pragma unroll
    for (int vi = 0; vi < 8; ++vi) {
        C[(crow0 + vi) * ldc + ccol0] = acc00[vi] + bv0;
        C[(crow0 + vi) * ldc + ccol1] = acc01[vi] + bv1;
        C[(crow1 + vi) * ldc + ccol0] = acc10[vi] + bv0;
        C[(crow1 + vi) * ldc + ccol1] = acc11[vi] + bv1;
    }
}

// ---------------------------------------------------------------------------
// GEMM1 (NT, B row-major KxN needs transpose-on-load): VGPR-staged path.
// Block = 128 thr (4 waves) -> 32x32 C tile.  Batched over blockIdx.z.
// ---------------------------------------------------------------------------
__global__ __launch_bounds__(128) void gemm_wmma_f32_nt_k(
    const float* __restrict__ A, const float* __restrict__ Bm,
    float* __restrict__ C,
    int K, int lda, int ldb, int ldc, int sB, int sC)
{
    __shared__ float lA[32][LPAD];
    __shared__ float lB[32][LPAD];

    const int tid  = threadIdx.x;
    const int lane = tid & 31;
    const int wave = tid >> 5;
    const int wm   = wave >> 1;
    const int wn   = wave & 1;
    const int half = lane >> 4;
    const int l16  = lane & 15;
    const int row0 = blockIdx.y * 32;
    const int col0 = blockIdx.x * 32;
    Bm += (size_t)blockIdx.z * sB;
    C  += (size_t)blockIdx.z * sC;

    const int lr = tid >> 3;          // 0..15
    const int lk = (tid & 7) << 2;    // 0,4,...,28

    v8f acc = {};
    for (int kc = 0; kc < K; kc += 32) {
        #pragma unroll
        for (int it = 0; it < 2; ++it) {
            const int r = lr + it * 16;
            const float4 va = *reinterpret_cast<const float4*>(
                &A[(row0 + r) * lda + kc + lk]);
            lA[r][lk + 0] = va.x; lA[r][lk + 1] = va.y;
            lA[r][lk + 2] = va.z; lA[r][lk + 3] = va.w;
            const int kr = r;
            const float4 vb = *reinterpret_cast<const float4*>(
                &Bm[(kc + kr) * ldb + col0 + lk]);
            lB[lk + 0][kr] = vb.x; lB[lk + 1][kr] = vb.y;
            lB[lk + 2][kr] = vb.z; lB[lk + 3][kr] = vb.w;
        }
        __syncthreads();
        const int ra = (wm * 16 + l16);
        const int cb = (wn * 16 + l16);
        #pragma unroll
        for (int kk = 0; kk < 32; kk += 4) {
            const float2 af = *reinterpret_cast<const float2*>(&lA[ra][kk + 2 * half]);
            const float2 bf = *reinterpret_cast<const float2*>(&lB[cb][kk + 2 * half]);
            v2f a; a.x = af.x; a.y = af.y;
            v2f b; b.x = bf.x; b.y = bf.y;
            acc = __builtin_amdgcn_wmma_f32_16x16x4_f32(false, a, false, b, (short)0, acc, false, false);
        }
        __syncthreads();
    }
    const int ccol = col0 + wn * 16 + l16;
    #pragma unroll
    for (int vi = 0; vi < 8; ++vi) {
        const int crow = row0 + wm * 16 + vi + 8 * half;
        C[crow * ldc + ccol] = acc[vi];
    }
}

// ---------------------------------------------------------------------------
// Attention: one wave32 per (b,n,h); lane owns 2 of 64 d-dims.
// ---------------------------------------------------------------------------
__global__ __launch_bounds__(256) void attention_k(
    const float* __restrict__ qb, const float* __restrict__ kv,
    const float* __restrict__ rel_bias, const uint8_t* __restrict__ mask,
    float* __restrict__ attn_out)
{
    const int wid  = blockIdx.x * 8 + (threadIdx.x >> 5);  // 0 .. B*N*H-1
    const int lane = threadIdx.x & 31;
    const int h    = wid & (HH - 1);
    const int bn   = wid >> 3;                             // b*128 + n
    const int n    = bn & (NN - 1);
    const int d0   = lane * 2;

    const float2 q2 = *reinterpret_cast<const float2*>(&qb[bn * CC + h * 64 + d0]);

    float logit[RR];
    #pragma unroll
    for (int r = 0; r < RR; ++r) {
        const int krow = bn * RR + r;
        const float2 k2 = *reinterpret_cast<const float2*>(&kv[krow * 1024 + h * 64 + d0]);
        float p = q2.x * k2.x + q2.y * k2.y;
        #pragma unroll
        for (int s = 16; s >= 1; s >>= 1) p += __shfl_xor(p, s, 32);
        float l = p * 0.125f + rel_bias[h * RR + r];
        if (mask[n * RR + r]) l = -__builtin_inff();
        logit[r] = l;
    }
    float mx = logit[0];
    #pragma unroll
    for (int r = 1; r < RR; ++r) mx = fmaxf(mx, logit[r]);
    float sum = 0.0f;
    #pragma unroll
    for (int r = 0; r < RR; ++r) { logit[r] = __expf(logit[r] - mx); sum += logit[r]; }
    const float inv = 1.0f / sum;

    float ox = 0.0f, oy = 0.0f;
    #pragma unroll
    for (int r = 0; r < RR; ++r) {
        const int vrow = bn * RR + r;
        const float2 v2 = *reinterpret_cast<const float2*>(&kv[vrow * 1024 + 512 + h * 64 + d0]);
        const float w = logit[r] * inv;
        ox += w * v2.x; oy += w * v2.y;
    }
    float2 o; o.x = ox; o.y = oy;
    *reinterpret_cast<float2*>(&attn_out[bn * CC + h * 64 + d0]) = o;
}

// ---------------------------------------------------------------------------
// Launcher
// ---------------------------------------------------------------------------
extern "C" void kernel_launch(void* const* d_in, const int* in_sizes, int n_in,
                              void* d_out, int out_size, void* d_ws, size_t ws_size,
                              hipStream_t stream)
{
    (void)in_sizes; (void)n_in; (void)out_size; (void)ws_size;
    const float*   x        = (const float*)d_in[0];   // (16,128,512)
    const float*   asg      = (const float*)d_in[1];   // (128,128,17)
    const uint8_t* mask     = (const uint8_t*)d_in[2]; // (128,17) bool (1B)
    const float*   q_w      = (const float*)d_in[3];   // (512,512)
    const float*   kv_w     = (const float*)d_in[4];   // (1024,512)
    const float*   rel_bias = (const float*)d_in[5];   // (8,1,17)
    const float*   proj_w   = (const float*)d_in[6];   // (512,512)
    const float*   proj_b   = (const float*)d_in[7];   // (512,)
    float* out = (float*)d_out;
    float* ws  = (float*)d_ws;

    // Workspace layout (floats). q/attn_out overlay pre_kv (dead after GEMM2).
    const long long OFF_AP    = 0;                        // 2176*128   = 278528
    const long long OFF_PREKV = 278528;                   // 34816*512  = 17825792
    const long long OFF_KV    = OFF_PREKV + 17825792LL;   // 34816*1024 = 35651584
    // total 53,755,904 floats = 215 MB
    float* Ap     = ws + OFF_AP;
    float* pre_kv = ws + OFF_PREKV;
    float* kvb    = ws + OFF_KV;
    float* qb     = ws + OFF_PREKV;                       // overlay
    float* attn_o = ws + OFF_PREKV + 1048576LL;           // overlay

    // 0) pack assignment -> Ap (2176 x 128)
    pack_assignment_k<<<(MR * NN + 255) / 256, 256, 0, stream>>>(asg, Ap);

    // 1) pre_kv[b] = Ap @ x[b]   (M=2176, N=512, K=128), batched over b
    gemm_wmma_f32_nt_k<<<dim3(CC / 32, MR / 32, BB), 128, 0, stream>>>(
        Ap, x, pre_kv, /*K=*/NN, /*lda=*/NN, /*ldb=*/CC, /*ldc=*/CC,
        /*sB=*/NN * CC, /*sC=*/MR * CC);

    // 2) kv = pre_kv @ kv_w^T   (M=34816, N=1024, K=512)  [TDM + WMMA]
    gemm_wmma_f32_tdm_k<false><<<dim3(1024 / 64, MROWS / 64), 128, 0, stream>>>(
        pre_kv, kv_w, nullptr, kvb, /*K=*/CC, CC, CC, 1024);

    // 3) q = x @ q_w^T          (M=2048, N=512, K=512)    [TDM + WMMA]
    gemm_wmma_f32_tdm_k<false><<<dim3(CC / 64, (BB * NN) / 64), 128, 0, stream>>>(
        x, q_w, nullptr, qb, /*K=*/CC, CC, CC, CC);

    // 4) attention: B*N*H = 16384 waves, 8 waves per 256-thread block
    attention_k<<<(BB * NN * HH) / 8, 256, 0, stream>>>(qb, kvb, rel_bias, mask, attn_o);

    // 5) out = attn_out @ proj_w^T + proj_b   (M=2048, N=512, K=512) [TDM + WMMA]
    gemm_wmma_f32_tdm_k<true><<<dim3(CC / 64, (BB * NN) / 64), 128, 0, stream>>>(
        attn_o, proj_w, proj_b, out, /*K=*/CC, CC, CC, CC);
}